// MFModel_55731495633505
// MI455X (gfx1250) — compile-verified
//
#include <hip/hip_runtime.h>

typedef float v2f __attribute__((ext_vector_type(2)));
typedef float v8f __attribute__((ext_vector_type(8)));

// One wave32 computes 16 edge dot-products (D=64) via 16 accumulating
// V_WMMA_F32_16X16X4_F32 ops; diagonal of the 16x16 result = per-edge scores.
__global__ __launch_bounds__(256) void edge_dot_wmma(
    const float* __restrict__ user_table,   // [1e6, 64]
    const float* __restrict__ item_table,   // [1e6, 64]
    const int*   __restrict__ idx_user,     // [NU]
    const int*   __restrict__ idx_item,     // [NI]
    const int*   __restrict__ pos_src,      // [E]
    const int*   __restrict__ pos_dst,      // [E]
    const int*   __restrict__ neg_src,      // [E]
    const int*   __restrict__ neg_dst,      // [E]
    float*       __restrict__ out,          // [2E] = pos_score | neg_score
    int E, int totalEdges)
{
    const int lane = threadIdx.x & 31;
    const int wave = threadIdx.x >> 5;
    const int tile = blockIdx.x * 8 + wave;      // 16 edges per tile
    const int r    = lane & 15;                  // edge-in-tile this lane serves
    const int hi   = lane >> 4;                  // K-half (0: K=0..1, 1: K=2..3)

    int e = tile * 16 + r;
    // Clamp (branch-free) so EXEC stays all-ones through the WMMA region.
    int ec = (e < totalEdges) ? e : (totalEdges - 1);

    const bool isPos = (ec < E);
    const int  eo    = isPos ? ec : (ec - E);
    const int* sp    = isPos ? pos_src : neg_src;
    const int* dp    = isPos ? pos_dst : neg_dst;

    const int srow = idx_user[sp[eo]];
    const int drow = idx_item[dp[eo]];

    const float* __restrict__ up = user_table + (size_t)srow * 64;
    const float* __restrict__ vp = item_table + (size_t)drow * 64;

    v8f c = {0.f, 0.f, 0.f, 0.f, 0.f, 0.f, 0.f, 0.f};
    const int o = 2 * hi;                        // lane's K offset inside K4 slice

    #pragma unroll
    for (int k = 0; k < 16; ++k) {
        // A(16x4) f32 layout: lanes 0-15 -> {K=0 (v0), K=1 (v1)},
        //                     lanes 16-31 -> {K=2 (v0), K=3 (v1)}.
        // B(4x16) mirrors this; both reduce to a float2 load at 4k + 2*hi.
        v2f a = *(const v2f*)(up + 4 * k + o);
        v2f b = *(const v2f*)(vp + 4 * k + o);
        // 8 args: (neg_a, A, neg_b, B, c_mod, C, reuse_a, reuse_b)
        c = __builtin_amdgcn_wmma_f32_16x16x4_f32(
                false, a, false, b, (short)0, c, false, false);
    }

    // Diagonal extraction from the 16x16 f32 C/D layout:
    //   m < 8 : value in VGPR c[m],   lane m
    //   m >= 8: value in VGPR c[m-8], lane m+16
    int sel = (lane < 8) ? lane : (lane - 24);   // only meaningful on active lanes
    float d = c[0];
    d = (sel == 1) ? c[1] : d;
    d = (sel == 2) ? c[2] : d;
    d = (sel == 3) ? c[3] : d;
    d = (sel == 4) ? c[4] : d;
    d = (sel == 5) ? c[5] : d;
    d = (sel == 6) ? c[6] : d;
    d = (sel == 7) ? c[7] : d;

    const bool active = (lane < 8) || (lane >= 24);
    if (active && e < totalEdges) {
        out[e] = d;                              // e == tile*16 + m on active lanes
    }
}

extern "C" void kernel_launch(void* const* d_in, const int* in_sizes, int n_in,
                              void* d_out, int out_size, void* d_ws, size_t ws_size,
                              hipStream_t stream) {
    const float* user_table = (const float*)d_in[0];
    const float* item_table = (const float*)d_in[1];
    const int*   idx_user   = (const int*)d_in[2];
    const int*   idx_item   = (const int*)d_in[3];
    const int*   pos_src    = (const int*)d_in[4];
    const int*   pos_dst    = (const int*)d_in[5];
    const int*   neg_src    = (const int*)d_in[6];
    const int*   neg_dst    = (const int*)d_in[7];
    float*       out        = (float*)d_out;

    const int E     = in_sizes[4];
    const int total = 2 * E;                     // pos scores then neg scores
    const int tiles = (total + 15) / 16;         // 16 edges per wave
    const int blocks = (tiles + 7) / 8;          // 8 waves per 256-thread block

    edge_dot_wmma<<<blocks, 256, 0, stream>>>(
        user_table, item_table, idx_user, idx_item,
        pos_src, pos_dst, neg_src, neg_dst, out, E, total);
}